// _GAT_30932354465910
// MI455X (gfx1250) — compile-verified
//
#include <hip/hip_runtime.h>

#define N_NODES 100000
#define IN_DIM  512
#define HIDDEN  64
#define OUT_DIM 32
#define N_EDGES 1600000
#define NEG_SLOPE 0.2f

typedef __attribute__((ext_vector_type(2))) float v2f;
typedef __attribute__((ext_vector_type(8))) float v8f;

// Monotone float<->ordered-int mapping so signed-int atomicMax == float max.
__device__ __forceinline__ int f2ord(float f) {
  int i = __float_as_int(f);
  return i >= 0 ? i : (i ^ 0x7fffffff);
}
__device__ __forceinline__ float ord2f(int i) {
  return __int_as_float(i >= 0 ? i : (i ^ 0x7fffffff));
}

__device__ __forceinline__ void atomic_add_f32(float* p, float v) {
  __hip_atomic_fetch_add(p, v, __ATOMIC_RELAXED, __HIP_MEMORY_SCOPE_AGENT);
}

// ---------------------------------------------------------------------------
// Transpose W1 [512,64] -> Wt1 [64,512] and W2 [64,32] -> Wt2 [32,64]
// (n-major so WMMA B fragments become contiguous v2f loads)
// ---------------------------------------------------------------------------
__global__ void k_transpose(const float* __restrict__ W1, const float* __restrict__ W2,
                            float* __restrict__ Wt1, float* __restrict__ Wt2) {
  int t = blockIdx.x * blockDim.x + threadIdx.x;
  if (t < IN_DIM * HIDDEN) {
    int k = t / HIDDEN, n = t % HIDDEN;
    Wt1[n * IN_DIM + k] = W1[t];
  }
  int t2 = t - IN_DIM * HIDDEN;
  if (t2 >= 0 && t2 < HIDDEN * OUT_DIM) {
    int k = t2 / OUT_DIM, n = t2 % OUT_DIM;
    Wt2[n * HIDDEN + k] = W2[t2];
  }
}

// ---------------------------------------------------------------------------
// GEMM1: H[100000,64] = X[100000,512] @ W1[512,64]  (fp32 WMMA 16x16x4)
// One wave -> 16 rows x 64 cols (4 col tiles); X streamed exactly once.
// ---------------------------------------------------------------------------
__global__ void k_gemm1(const float* __restrict__ X, const float* __restrict__ Wt1,
                        float* __restrict__ H) {
  int wave = blockIdx.x * (blockDim.x >> 5) + (threadIdx.x >> 5);
  int lane = threadIdx.x & 31;
  int row0 = wave * 16;               // exact: 6250 waves * 16 = 100000
  int rA   = row0 + (lane & 15);      // A-matrix row for this lane
  int kk   = (lane >> 4) * 2;         // K offset: lanes 0-15 -> K+0,1; 16-31 -> K+2,3
  const float* xrow = X   + (size_t)rA * IN_DIM + kk;
  const float* b0   = Wt1 + (size_t)(lane & 15) * IN_DIM + kk;
  v8f acc0 = {}, acc1 = {}, acc2 = {}, acc3 = {};
  for (int k = 0; k < IN_DIM; k += 4) {
    v2f a   = *(const v2f*)(xrow + k);
    v2f bb0 = *(const v2f*)(b0 + k);
    v2f bb1 = *(const v2f*)(b0 + 16 * IN_DIM + k);
    v2f bb2 = *(const v2f*)(b0 + 32 * IN_DIM + k);
    v2f bb3 = *(const v2f*)(b0 + 48 * IN_DIM + k);
    acc0 = __builtin_amdgcn_wmma_f32_16x16x4_f32(false, a, false, bb0, (short)0, acc0, false, false);
    acc1 = __builtin_amdgcn_wmma_f32_16x16x4_f32(false, a, false, bb1, (short)0, acc1, false, false);
    acc2 = __builtin_amdgcn_wmma_f32_16x16x4_f32(false, a, false, bb2, (short)0, acc2, false, false);
    acc3 = __builtin_amdgcn_wmma_f32_16x16x4_f32(false, a, false, bb3, (short)0, acc3, false, false);
  }
  int mOff = (lane >> 4) * 8;         // C/D: lanes 16-31 hold M = r+8
  int col  = lane & 15;
  float* hout = H + (size_t)row0 * HIDDEN + col;
#pragma unroll
  for (int r = 0; r < 8; ++r) {
    size_t ro = (size_t)(r + mOff) * HIDDEN;
    hout[ro +  0] = acc0[r];
    hout[ro + 16] = acc1[r];
    hout[ro + 32] = acc2[r];
    hout[ro + 48] = acc3[r];
  }
}

// ---------------------------------------------------------------------------
// Per-node: s = h.a_src, d = h.a_dst; init segment-max / denom / accumulator
// ---------------------------------------------------------------------------
__global__ void k_nodes(const float* __restrict__ H, const float* __restrict__ Asrc,
                        const float* __restrict__ Adst, float* __restrict__ S,
                        float* __restrict__ D, int* __restrict__ Mord,
                        float* __restrict__ Den, float* __restrict__ OUTA) {
  int i = blockIdx.x * blockDim.x + threadIdx.x;
  if (i >= N_NODES) return;
  const float* hr = H + (size_t)i * HIDDEN;
  float* oz = OUTA + (size_t)i * HIDDEN;
  float s = 0.f, d = 0.f;
#pragma unroll 8
  for (int j = 0; j < HIDDEN; ++j) {
    float hv = hr[j];
    s += hv * Asrc[j];
    d += hv * Adst[j];
    oz[j] = 0.f;
  }
  S[i] = s;
  D[i] = d;
  Mord[i] = f2ord(-__builtin_inff());
  Den[i] = 0.f;
}

// ---------------------------------------------------------------------------
// Edge pass A: e = LeakyReLU(s[src]+d[dst]); segment max via ordered-int atomicMax
// ---------------------------------------------------------------------------
__global__ void k_edge_max(const int* __restrict__ SRC, const int* __restrict__ DST,
                           const float* __restrict__ S, const float* __restrict__ D,
                           float* __restrict__ Ebuf, int* __restrict__ Mord) {
  int e = blockIdx.x * blockDim.x + threadIdx.x;
  if (e >= N_EDGES) return;
  int s = SRC[e], d = DST[e];
  float v = S[s] + D[d];
  v = v > 0.f ? v : NEG_SLOPE * v;
  Ebuf[e] = v;
  atomicMax(&Mord[d], f2ord(v));
}

// ---------------------------------------------------------------------------
// Edge pass B (fused): ex = exp(e - m[dst]); denom[dst] += ex;
//                      OUTA[dst] += ex * h[src]   (normalization deferred)
// 16 lanes per edge, float4 per lane -> one coalesced 256B h[src] gather.
// ---------------------------------------------------------------------------
__global__ void k_edge_acc(const int* __restrict__ SRC, const int* __restrict__ DST,
                           const float* __restrict__ Ebuf, const int* __restrict__ Mord,
                           const float* __restrict__ H, float* __restrict__ Den,
                           float* __restrict__ OUTA) {
  int gi = blockIdx.x * blockDim.x + threadIdx.x;
  int e = gi >> 4;
  int part = gi & 15;
  if (e >= N_EDGES) return;
  int sN = SRC[e], dN = DST[e];
  float ex = expf(Ebuf[e] - ord2f(Mord[dN]));
  if (part == 0) atomic_add_f32(&Den[dN], ex);
  const float* hs = H + (size_t)sN * HIDDEN + part * 4;
  float* op = OUTA + (size_t)dN * HIDDEN + part * 4;
  float4 hv = *(const float4*)hs;
  atomic_add_f32(op + 0, ex * hv.x);
  atomic_add_f32(op + 1, ex * hv.y);
  atomic_add_f32(op + 2, ex * hv.z);
  atomic_add_f32(op + 3, ex * hv.w);
}

// ---------------------------------------------------------------------------
// GEMM2: Y[100000,32] = ELU(OUTA/denom) @ W2[64,32]  (fp32 WMMA 16x16x4)
// ELU + normalization applied while loading A fragments.
// ---------------------------------------------------------------------------
__global__ void k_gemm2(const float* __restrict__ OUTA, const float* __restrict__ Den,
                        const float* __restrict__ Wt2, float* __restrict__ Y) {
  int wave = blockIdx.x * (blockDim.x >> 5) + (threadIdx.x >> 5);
  int lane = threadIdx.x & 31;
  int row0 = wave * 16;
  int rA   = row0 + (lane & 15);
  int kk   = (lane >> 4) * 2;
  float rden = 1.0f / (Den[rA] + 1e-16f);
  const float* arow = OUTA + (size_t)rA * HIDDEN + kk;
  const float* b0   = Wt2 + (size_t)(lane & 15) * HIDDEN + kk;
  v8f acc0 = {}, acc1 = {};
  for (int k = 0; k < HIDDEN; k += 4) {
    v2f a = *(const v2f*)(arow + k);
    a.x *= rden;
    a.y *= rden;
    a.x = a.x > 0.f ? a.x : (expf(a.x) - 1.f);   // ELU(alpha=1)
    a.y = a.y > 0.f ? a.y : (expf(a.y) - 1.f);
    v2f bb0 = *(const v2f*)(b0 + k);
    v2f bb1 = *(const v2f*)(b0 + 16 * HIDDEN + k);
    acc0 = __builtin_amdgcn_wmma_f32_16x16x4_f32(false, a, false, bb0, (short)0, acc0, false, false);
    acc1 = __builtin_amdgcn_wmma_f32_16x16x4_f32(false, a, false, bb1, (short)0, acc1, false, false);
  }
  int mOff = (lane >> 4) * 8;
  int col  = lane & 15;
  float* yout = Y + (size_t)row0 * OUT_DIM + col;
#pragma unroll
  for (int r = 0; r < 8; ++r) {
    size_t ro = (size_t)(r + mOff) * OUT_DIM;
    yout[ro +  0] = acc0[r];
    yout[ro + 16] = acc1[r];
  }
}

// ---------------------------------------------------------------------------
extern "C" void kernel_launch(void* const* d_in, const int* in_sizes, int n_in,
                              void* d_out, int out_size, void* d_ws, size_t ws_size,
                              hipStream_t stream) {
  (void)in_sizes; (void)n_in; (void)out_size; (void)ws_size;
  const float* X    = (const float*)d_in[0];
  const int*   EI   = (const int*)d_in[1];     // edge_index [2, E]
  const float* W1   = (const float*)d_in[2];
  const float* Asrc = (const float*)d_in[3];
  const float* Adst = (const float*)d_in[4];
  const float* W2   = (const float*)d_in[5];
  float* Y = (float*)d_out;

  float* ws   = (float*)d_ws;
  float* Wt1  = ws;                                  // 64*512
  float* Wt2  = Wt1 + IN_DIM * HIDDEN;               // 32*64
  float* H    = Wt2 + HIDDEN * OUT_DIM;              // 100000*64
  float* S    = H + (size_t)N_NODES * HIDDEN;        // 100000
  float* D    = S + N_NODES;                         // 100000
  int*   Mord = (int*)(D + N_NODES);                 // 100000
  float* Den  = (float*)(Mord + N_NODES);            // 100000
  float* OUTA = Den + N_NODES;                       // 100000*64
  float* Ebuf = OUTA + (size_t)N_NODES * HIDDEN;     // 1600000

  const int* SRC = EI;
  const int* DST = EI + N_EDGES;

  k_transpose<<<(IN_DIM * HIDDEN + HIDDEN * OUT_DIM + 255) / 256, 256, 0, stream>>>(W1, W2, Wt1, Wt2);
  k_gemm1<<<625, 320, 0, stream>>>(X, Wt1, H);                       // 6250 waves = 6250 row tiles
  k_nodes<<<(N_NODES + 255) / 256, 256, 0, stream>>>(H, Asrc, Adst, S, D, Mord, Den, OUTA);
  k_edge_max<<<(N_EDGES + 255) / 256, 256, 0, stream>>>(SRC, DST, S, D, Ebuf, Mord);
  k_edge_acc<<<(N_EDGES * 16) / 256, 256, 0, stream>>>(SRC, DST, Ebuf, Mord, H, Den, OUTA);
  k_gemm2<<<625, 320, 0, stream>>>(OUTA, Den, Wt2, Y);
}